// GraphSAGENet_8830452761405
// MI455X (gfx1250) — compile-verified
//
#include <hip/hip_runtime.h>
#include <hip/hip_bf16.h>

// ---------------------------------------------------------------------------
// GraphSAGE (L=4, D=128, OUT=26) for gfx1250 / MI455X.
// fp32 end-to-end using V_WMMA_F32_16X16X4_F32 (wave32 matrix pipe).
// ---------------------------------------------------------------------------

#define D_FEAT 128
#define LDS_STRIDE 132   // 128 + 4 pad: (4*row + k) mod 64 unique per row -> no bank conflicts
#define OUT_DIM 26

typedef __attribute__((ext_vector_type(2))) float v2f;
typedef __attribute__((ext_vector_type(8))) float v8f;

// ---------------- utility kernels ----------------

__global__ void sage_zero_f32(float* __restrict__ p, int n) {
    int i = blockIdx.x * blockDim.x + threadIdx.x;
    if (i < n) p[i] = 0.0f;
}

__global__ void sage_deg_kernel(const int* __restrict__ dst, float* __restrict__ deg,
                                int n_edges) {
    int e = blockIdx.x * blockDim.x + threadIdx.x;
    if (e < n_edges) atomicAdd(&deg[dst[e]], 1.0f);
}

__global__ void sage_invdeg_kernel(float* __restrict__ deg, int n_nodes) {
    int i = blockIdx.x * blockDim.x + threadIdx.x;
    if (i < n_nodes) {
        float d = deg[i];
        deg[i] = 1.0f / fmaxf(d, 1.0f);
    }
}

// One edge handled by 32 threads; each thread moves 4 floats (float4 gather,
// 4x global_atomic_add_f32 scatter). x (51 MB) and msg live in the 192 MB L2.
__global__ void sage_scatter_kernel(const int* __restrict__ src, const int* __restrict__ dst,
                                    const float* __restrict__ x, float* __restrict__ msg,
                                    int n_edges) {
    long long tid = (long long)blockIdx.x * blockDim.x + threadIdx.x;
    int e = (int)(tid >> 5);
    int c = (int)(tid & 31);
    if (e >= n_edges) return;
    int s = src[e];
    int d = dst[e];
    const float4 v = *(const float4*)(x + (size_t)s * D_FEAT + c * 4);
    float* p = msg + (size_t)d * D_FEAT + c * 4;
    atomicAdd(p + 0, v.x);
    atomicAdd(p + 1, v.y);
    atomicAdd(p + 2, v.z);
    atomicAdd(p + 3, v.w);
}

// ---------------- fused layer GEMM: relu(x@Ws + (msg*invdeg)@Wn + b) ----------------
// Block = one 16-row M tile, 256 threads = 8 waves; wave w owns N cols [16w,16w+16).
// A tiles staged in LDS (padded stride), weights stream from L2 (64 KB each, hot).
__global__ void __launch_bounds__(256)
sage_layer_kernel(const float* __restrict__ xin, const float* __restrict__ msg,
                  const float* __restrict__ invdeg,
                  const float* __restrict__ Wself, const float* __restrict__ Wneigh,
                  const float* __restrict__ bias, float* __restrict__ xout) {
    __shared__ float As[16 * LDS_STRIDE];
    __shared__ float Hs[16 * LDS_STRIDE];

    const int m0 = blockIdx.x * 16;
    const int t  = threadIdx.x;

    // ---- stage 16x128 x-tile and scaled msg-tile: 8 floats per thread each ----
    {
        int r  = t >> 4;          // 0..15
        int c0 = (t & 15) * 8;    // 0,8,...,120
        const float* xr = xin + (size_t)(m0 + r) * D_FEAT + c0;
        float4 a0 = *(const float4*)(xr);
        float4 a1 = *(const float4*)(xr + 4);
        *(float4*)(As + r * LDS_STRIDE + c0)     = a0;
        *(float4*)(As + r * LDS_STRIDE + c0 + 4) = a1;

        float iv = invdeg[m0 + r];
        const float* mr = msg + (size_t)(m0 + r) * D_FEAT + c0;
        float4 h0 = *(const float4*)(mr);
        float4 h1 = *(const float4*)(mr + 4);
        h0.x *= iv; h0.y *= iv; h0.z *= iv; h0.w *= iv;
        h1.x *= iv; h1.y *= iv; h1.z *= iv; h1.w *= iv;
        *(float4*)(Hs + r * LDS_STRIDE + c0)     = h0;
        *(float4*)(Hs + r * LDS_STRIDE + c0 + 4) = h1;
    }
    __syncthreads();

    const int wave = t >> 5;          // N-tile id 0..7
    const int lane = t & 31;
    const int n0   = wave * 16;
    const int half = lane >> 4;       // 0: K{0,1}, 1: K{2,3} (A 16x4 f32 layout)
    const int l16  = lane & 15;
    const int col  = n0 + l16;

    v8f acc;
    {
        float bv = bias[col];         // per-column bias, same for all 8 C rows
        #pragma unroll
        for (int i = 0; i < 8; ++i) acc[i] = bv;
    }

    #pragma unroll 4
    for (int k0 = 0; k0 < D_FEAT; k0 += 4) {
        const int ka = k0 + half * 2;
        // A fragment: row l16, K = ka, ka+1   (LDS, 8B aligned)
        v2f a = *(const v2f*)(As + l16 * LDS_STRIDE + ka);
        // B fragment: rows K=ka,ka+1 of column `col` (row-major 128x128 weight)
        v2f b;
        b.x = Wself[(size_t)ka * D_FEAT + col];
        b.y = Wself[(size_t)(ka + 1) * D_FEAT + col];
        acc = __builtin_amdgcn_wmma_f32_16x16x4_f32(false, a, false, b,
                                                    (short)0, acc, false, false);

        v2f h = *(const v2f*)(Hs + l16 * LDS_STRIDE + ka);
        v2f bn;
        bn.x = Wneigh[(size_t)ka * D_FEAT + col];
        bn.y = Wneigh[(size_t)(ka + 1) * D_FEAT + col];
        acc = __builtin_amdgcn_wmma_f32_16x16x4_f32(false, h, false, bn,
                                                    (short)0, acc, false, false);
    }

    // C/D layout: VGPR v -> row (v + 8*half), lanes l16 -> column
    #pragma unroll
    for (int v = 0; v < 8; ++v) {
        int row = m0 + v + half * 8;
        float val = acc[v];
        val = val > 0.0f ? val : 0.0f;            // ReLU
        xout[(size_t)row * D_FEAT + col] = val;
    }
}

// ---------------- prediction head: x @ W_pred + b_pred  (N = 26) ----------------
// 2 waves per 16-row tile: N-tile 0 = cols 0..15, N-tile 1 = cols 16..31 (masked).
__global__ void __launch_bounds__(64)
sage_pred_kernel(const float* __restrict__ xin, const float* __restrict__ Wp,
                 const float* __restrict__ bp, float* __restrict__ out) {
    const int m0   = blockIdx.x * 16;
    const int t    = threadIdx.x;     // 0..63
    const int wave = t >> 5;
    const int lane = t & 31;
    const int n0   = wave * 16;
    const int half = lane >> 4;
    const int l16  = lane & 15;
    const int col  = n0 + l16;
    const int bcol = col < OUT_DIM ? col : (OUT_DIM - 1);  // clamp (garbage lanes never store)

    v8f acc;
    {
        float bv = bp[bcol];
        #pragma unroll
        for (int i = 0; i < 8; ++i) acc[i] = bv;
    }

    #pragma unroll 4
    for (int k0 = 0; k0 < D_FEAT; k0 += 4) {
        const int ka = k0 + half * 2;
        const float* ap = xin + (size_t)(m0 + l16) * D_FEAT + ka;
        v2f a; a.x = ap[0]; a.y = ap[1];
        v2f b;
        b.x = Wp[(size_t)ka * OUT_DIM + bcol];
        b.y = Wp[(size_t)(ka + 1) * OUT_DIM + bcol];
        acc = __builtin_amdgcn_wmma_f32_16x16x4_f32(false, a, false, b,
                                                    (short)0, acc, false, false);
    }

    if (col < OUT_DIM) {
        #pragma unroll
        for (int v = 0; v < 8; ++v) {
            int row = m0 + v + half * 8;
            out[(size_t)row * OUT_DIM + col] = acc[v];
        }
    }
}

// ---------------------------------------------------------------------------

extern "C" void kernel_launch(void* const* d_in, const int* in_sizes, int n_in,
                              void* d_out, int out_size, void* d_ws, size_t ws_size,
                              hipStream_t stream) {
    const float* x       = (const float*)d_in[0];
    const int*   src     = (const int*)  d_in[1];
    const int*   dst     = (const int*)  d_in[2];
    const float* W_self  = (const float*)d_in[3];
    const float* W_neigh = (const float*)d_in[4];
    const float* b       = (const float*)d_in[5];
    const float* W_pred  = (const float*)d_in[6];
    const float* b_pred  = (const float*)d_in[7];

    const int n_nodes = in_sizes[0] / D_FEAT;
    const int n_edges = in_sizes[1];
    const int L       = in_sizes[5] / D_FEAT;
    const int feat_elems = n_nodes * D_FEAT;

    // workspace layout: invdeg | msg | bufA | bufB
    char*  ws = (char*)d_ws;
    size_t off = ((size_t)n_nodes * sizeof(float) + 511) & ~(size_t)511;
    size_t feat_bytes = (size_t)feat_elems * sizeof(float);
    float* invdeg = (float*)ws;
    float* msg    = (float*)(ws + off);
    float* bufA   = (float*)(ws + off + feat_bytes);
    float* bufB   = (float*)(ws + off + 2 * feat_bytes);

    const int T = 256;

    // ---- degrees -> reciprocal mean scale ----
    sage_zero_f32<<<(n_nodes + T - 1) / T, T, 0, stream>>>(invdeg, n_nodes);
    sage_deg_kernel<<<(n_edges + T - 1) / T, T, 0, stream>>>(dst, invdeg, n_edges);
    sage_invdeg_kernel<<<(n_nodes + T - 1) / T, T, 0, stream>>>(invdeg, n_nodes);

    const float* cur = x;
    float* nxt = bufA;
    const long long scat_threads = (long long)n_edges * 32;
    const int scat_blocks = (int)((scat_threads + T - 1) / T);
    const int m_tiles = n_nodes / 16;

    for (int l = 0; l < L; ++l) {
        sage_zero_f32<<<(feat_elems + T - 1) / T, T, 0, stream>>>(msg, feat_elems);
        sage_scatter_kernel<<<scat_blocks, T, 0, stream>>>(src, dst, cur, msg, n_edges);
        sage_layer_kernel<<<m_tiles, 256, 0, stream>>>(
            cur, msg, invdeg,
            W_self  + (size_t)l * D_FEAT * D_FEAT,
            W_neigh + (size_t)l * D_FEAT * D_FEAT,
            b       + (size_t)l * D_FEAT,
            nxt);
        cur = nxt;
        nxt = (nxt == bufA) ? bufB : bufA;
    }

    sage_pred_kernel<<<m_tiles, 64, 0, stream>>>(cur, W_pred, b_pred, (float*)d_out);
}